// LOCA_43843026158280
// MI455X (gfx1250) — compile-verified
//
#include <hip/hip_runtime.h>
#include <hip/hip_bf16.h>

typedef __attribute__((ext_vector_type(2))) float v2f;
typedef __attribute__((ext_vector_type(8))) float v8f;

#define NO 3
#define KD 3
#define C 256
#define H 64
#define W 64
#define BS 8
#define STEPS 3

// Per (pixel, channel): 3x3 depthwise correlation for 3 objects, softmax over
// objects, softmax-weighted sum. Branchless boundary handling (clamped address
// * mask) so EXEC stays full for the WMMA in the caller. Softmax normalizer
// uses hardware v_rcp_f32 (~1 ulp) instead of the IEEE divide sequence.
__device__ __forceinline__ float chan_val(const float* __restrict__ fimg,
                                          const float* __restrict__ sP,
                                          int c, int row, int x) {
    float n[9];
#pragma unroll
    for (int dy = 0; dy < 3; ++dy) {
#pragma unroll
        for (int dx = 0; dx < 3; ++dx) {
            int yy = row + dy - 1;
            int xx = x + dx - 1;
            float msk = ((unsigned)yy < 64u && (unsigned)xx < 64u) ? 1.0f : 0.0f;
            int yc = min(max(yy, 0), 63);
            int xc = min(max(xx, 0), 63);
            n[dy * 3 + dx] = fimg[yc * W + xc] * msk;
        }
    }
    float r0 = 0.f, r1 = 0.f, r2 = 0.f;
#pragma unroll
    for (int t = 0; t < 9; ++t) {
        float nv = n[t];
        r0 = fmaf(nv, sP[(0 * 9 + t) * C + c], r0);
        r1 = fmaf(nv, sP[(1 * 9 + t) * C + c], r1);
        r2 = fmaf(nv, sP[(2 * 9 + t) * C + c], r2);
    }
    float m  = fmaxf(r0, fmaxf(r1, r2));
    float e0 = __expf(r0 - m), e1 = __expf(r1 - m), e2 = __expf(r2 - m);
    float inv = __builtin_amdgcn_rcpf(e0 + e1 + e2);   // v_rcp_f32
    return (e0 * r0 + e1 * r1 + e2 * r2) * inv;
}

// Block = 256 threads (8 waves). Block covers 2 image rows of one (step,batch).
// Wave w: row = 2*blockIdx.x + w/4, x-segment = 16*(w%4). Each wave owns 16
// pixels (M of the WMMA tile); lane halves carry interleaved channel pairs
// matching the 16x16x4 f32 A-matrix layout (lanes 0-15: K0,K1; 16-31: K2,K3).
__global__ __launch_bounds__(256) void loca_fused_dmap(
    const float* __restrict__ f_e,      // [BS, C, H, W]
    const float* __restrict__ protos,   // [STEPS, NO*KD*KD, BS, C]
    const float* __restrict__ w_head,   // [C]
    const float* __restrict__ b_head,   // [1]
    float* __restrict__ dmap)           // [STEPS, BS, H, W]
{
    __shared__ float sP[NO * KD * KD * C];  // 27.6 KB
    __shared__ float sW[C];

    const int tid = threadIdx.x;
    const int b = blockIdx.y;
    const int s = blockIdx.z;

    // Stage prototypes for this (s,b): element [s, j, b, c].
    for (int i = tid; i < NO * KD * KD * C; i += 256) {
        int j = i >> 8;
        int c = i & (C - 1);
        sP[i] = protos[(((size_t)s * (NO * KD * KD) + j) * BS + b) * C + c];
    }
    sW[tid] = w_head[tid];
    __syncthreads();

    const int wave = tid >> 5;
    const int lane = tid & 31;
    const int row  = blockIdx.x * 2 + (wave >> 2);
    const int xseg = (wave & 3) << 4;
    const int x    = xseg + (lane & 15);
    const int half = lane >> 4;

    const float* fb  = f_e + (size_t)b * C * H * W;
    const v2f*   sW2 = (const v2f*)sW;      // pair view: guaranteed b64 LDS loads
    const float  bh  = b_head[0];           // hoisted: kmcnt wait overlaps loop

    v8f acc = {};
#pragma unroll 2
    for (int base = 0; base < C; base += 4) {
        int c0 = base + half * 2;
        // Issue the B-fragment LDS load FIRST so its dscnt wait is hidden
        // behind the long chan_val FMA/exp chains.
        v2f wv = sW2[c0 >> 1];              // {w[c0], w[c0+1]}, replicated over N
        float sA = chan_val(fb + (size_t)c0 * (H * W), sP, c0, row, x);
        float sB = chan_val(fb + (size_t)(c0 + 1) * (H * W), sP, c0 + 1, row, x);
        v2f a; a.x = sA; a.y = sB;
        // D[M,N] += sum_k A[M,k] * B[k,N]; every column holds the same dot.
        acc = __builtin_amdgcn_wmma_f32_16x16x4_f32(
            false, a, false, wv, (short)0, acc, false, false);
    }

    // C/D layout: VGPR r <-> M = r (lanes 0-15) or M = 8+r (lanes 16-31).
    const int r = lane & 7;
    float v = acc[0];
    v = (r == 1) ? acc[1] : v;
    v = (r == 2) ? acc[2] : v;
    v = (r == 3) ? acc[3] : v;
    v = (r == 4) ? acc[4] : v;
    v = (r == 5) ? acc[5] : v;
    v = (r == 6) ? acc[6] : v;
    v = (r == 7) ? acc[7] : v;

    if ((lane & 15) < 8) {           // lanes 0-7 and 16-23 each own one pixel
        int m = r + (half << 3);
        float outv = fmaxf(v + bh, 0.0f);
        dmap[(((size_t)s * BS + b) * H + row) * W + xseg + m] = outv;
    }
}

// Bilinear 8x upsample with half-pixel centers (matches jax.image.resize).
__global__ __launch_bounds__(256) void upsample8_bilinear(
    const float* __restrict__ dmap,   // [STEPS*BS, 64, 64]
    float* __restrict__ out)          // [STEPS, BS, 1, 512, 512]
{
    int idx = blockIdx.x * 256 + threadIdx.x;     // exactly STEPS*BS*512*512
    int X  = idx & 511;
    int Y  = (idx >> 9) & 511;
    int sb = idx >> 18;                           // 0..23

    float sx = (X + 0.5f) * 0.125f - 0.5f;
    float sy = (Y + 0.5f) * 0.125f - 0.5f;
    float fx0 = floorf(sx), fy0 = floorf(sy);
    int x0 = (int)fx0, y0 = (int)fy0;
    float tx = sx - fx0, ty = sy - fy0;
    int x0c = min(max(x0, 0), 63), x1c = min(max(x0 + 1, 0), 63);
    int y0c = min(max(y0, 0), 63), y1c = min(max(y0 + 1, 0), 63);

    const float* dm = dmap + (size_t)sb * (H * W);
    float v00 = dm[y0c * W + x0c], v01 = dm[y0c * W + x1c];
    float v10 = dm[y1c * W + x0c], v11 = dm[y1c * W + x1c];
    float top = v00 + (v01 - v00) * tx;
    float bot = v10 + (v11 - v10) * tx;
    out[idx] = top + (bot - top) * ty;
}

extern "C" void kernel_launch(void* const* d_in, const int* in_sizes, int n_in,
                              void* d_out, int out_size, void* d_ws, size_t ws_size,
                              hipStream_t stream) {
    const float* f_e    = (const float*)d_in[0];
    const float* protos = (const float*)d_in[1];
    const float* w_head = (const float*)d_in[2];
    const float* b_head = (const float*)d_in[3];
    float* out  = (float*)d_out;
    float* dmap = (float*)d_ws;   // STEPS*BS*64*64 floats = 384 KB scratch

    dim3 g1(32, BS, STEPS);       // 32 row-pairs x 8 batches x 3 steps
    loca_fused_dmap<<<g1, 256, 0, stream>>>(f_e, protos, w_head, b_head, dmap);

    int total = STEPS * BS * 512 * 512;           // 6,291,456
    upsample8_bilinear<<<total / 256, 256, 0, stream>>>(dmap, out);
}